// StructNDeconv2d_gy_with_ds_83502754169201
// MI455X (gfx1250) — compile-verified
//
#include <hip/hip_runtime.h>
#include <math.h>

// Problem constants (fixed by the reference)
#define Bb 2
#define Cc 32
#define Hh 176
#define Ww 608
#define HO 351          // (H-1)*2 - 2*1 + 3
#define WO 1215         // (W-1)*2 - 2*1 + 3
#define EPSC 1e-20f

// Tiling: each block produces a 32x64 output tile of one (b,c) plane.
#define TOH 32
#define TOW 64
#define TIH 17          // core input rows   = TOH/2 + 1
#define TIW 33          // core input cols   = TOW/2 + 1
#define SDH 19          // staged d/cd rows  = TIH + 2 (for +-1 H-roll)
#define TILES_Y 11      // ceil(351/32)
#define TILES_X 19      // ceil(1215/64)
#define NTHREADS 256

#ifndef __has_builtin
#define __has_builtin(x) 0
#endif

#if __has_builtin(__builtin_amdgcn_global_load_async_to_lds_b32)
#define HAVE_ASYNC 1
typedef __attribute__((address_space(1))) int g_i32;
typedef __attribute__((address_space(3))) int l_i32;
#else
#define HAVE_ASYNC 0
#endif

__device__ __forceinline__ float softplus_f(float x) {
    // numerically stable log1p(exp(x))
    return fmaxf(x, 0.0f) + log1pf(expf(-fabsf(x)));
}

__global__ __launch_bounds__(NTHREADS)
void fused_structn_deconv(const float* __restrict__ d,
                          const float* __restrict__ cd,
                          const float* __restrict__ gy,
                          const float* __restrict__ cgy,
                          const float* __restrict__ w_prop_raw,
                          const float* __restrict__ sw_raw,
                          const float* __restrict__ bias,
                          float* __restrict__ out) {
    __shared__ float sD [SDH * TIW];
    __shared__ float sCD[SDH * TIW];
    __shared__ float sNom[TIH * TIW];
    __shared__ float sCf [TIH * TIW];
    __shared__ float sPar[11];          // 9 softplus'd 3x3 weights, w_prop, bias

    const int tid = threadIdx.x;
    int t = blockIdx.x;
    const int tx = t % TILES_X; t /= TILES_X;
    const int ty = t % TILES_Y; t /= TILES_Y;
    const int p  = t;                   // plane = b*C + c, 0..63
    const int c  = p & (Cc - 1);

    const size_t planeOff = (size_t)p * (Hh * Ww);
    const float* dP  = d   + planeOff;
    const float* cdP = cd  + planeOff;
    const float* gyP = gy  + planeOff;
    const float* cgP = cgy + planeOff;

    const int r0 = ty * (TOH / 2);      // input-row base of core tile
    const int c0 = tx * (TOW / 2);      // input-col base of core tile

    // ---- Stage d / cd tiles into LDS (async gather), with jnp.roll H-wrap ----
    for (int i = tid; i < SDH * TIW; i += NTHREADS) {
        const int j = i / TIW;
        const int x = i - j * TIW;
        int gr = r0 - 1 + j;                    // wants rows r0-1 .. r0+17
        gr += (gr < 0)   ? Hh : 0;              // roll wrap (row -1 -> H-1)
        gr -= (gr >= Hh) ? Hh : 0;              // roll wrap (row H  -> 0)
        int gc = c0 + x;
        gc = (gc < Ww) ? gc : (Ww - 1);         // clamp (clamped lanes unused)
        const size_t off = (size_t)gr * Ww + gc;
#if HAVE_ASYNC
        __builtin_amdgcn_global_load_async_to_lds_b32(
            (g_i32*)(dP + off),  (l_i32*)(&sD[i]),  0, 0);
        __builtin_amdgcn_global_load_async_to_lds_b32(
            (g_i32*)(cdP + off), (l_i32*)(&sCD[i]), 0, 0);
#else
        sD[i]  = dP[off];
        sCD[i] = cdP[off];
#endif
    }

    // ---- Per-channel params (overlaps with async staging) ----
    if (tid < 9)            sPar[tid] = softplus_f(sw_raw[c * 9 + tid]);
    else if (tid == 9)      sPar[9]   = softplus_f(w_prop_raw[c]);
    else if (tid == 10)     sPar[10]  = bias[c];

#if HAVE_ASYNC
#if __has_builtin(__builtin_amdgcn_s_wait_asynccnt)
    __builtin_amdgcn_s_wait_asynccnt(0);
#else
    asm volatile("s_wait_asynccnt 0" ::: "memory");
#endif
#endif
    __syncthreads();

    float swv[9];
#pragma unroll
    for (int i = 0; i < 9; ++i) swv[i] = sPar[i];
    const float wp      = sPar[9];
    const float bs      = sPar[10];
    const float inv_wp1 = 1.0f / (wp + 1.0f);

    // ---- Phase 1: fused pre-processing into LDS (one rcp per input pixel) ----
    for (int i = tid; i < TIH * TIW; i += NTHREADS) {
        const int ly = i / TIW;
        const int x  = i - ly * TIW;
        const int ih = r0 + ly;
        const int iw = c0 + x;
        float nomv = 0.0f, cf = 0.0f;
        if (ih < Hh && iw < Ww) {
            const int ja = ly * TIW + x;          // staged row for ih-1
            const int jc = (ly + 2) * TIW + x;    // staged row for ih+1
            const float d_u   = sD[ja];
            const float d_dn  = sD[jc];
            const float cd_u  = (ih == 0)      ? 0.0f : sCD[ja];
            const float cd_dn = (ih == Hh - 1) ? 0.0f : sCD[jc];
            const float g  = gyP[(size_t)ih * Ww + iw];
            const float cg = cgP[(size_t)ih * Ww + iw];
            const float cfds = cd_u * cd_u;                       // cgy_from_ds
            const float hden = cd_u * d_u + cd_dn * d_dn;
            // (d_down - d_up)/2/height == (d_dn-d_u)*0.5*(cd_u+cd_dn)/hden
            const float gyds = (d_dn - d_u) * 0.5f * (cd_u + cd_dn) / hden;
            const float wcg  = wp * cg;
            const float num  = wcg * g + cfds * gyds;
            const float den  = wcg + cfds;
            cf   = den * inv_wp1;                 // cgy_f
            nomv = num * inv_wp1;                 // cgy_f * gy_f
        }
        sNom[i] = nomv;
        sCf[i]  = cf;
    }
    __syncthreads();

    // ---- Phase 2: stride-2 transposed 3x3 depthwise conv (all taps in-bounds) ----
    const int oh0 = ty * TOH, ow0 = tx * TOW;
    const size_t outPlane = (size_t)p * HO * WO;
    float* __restrict__ gyOut = out + outPlane;
    float* __restrict__ cgOut = out + (size_t)Bb * Cc * HO * WO + outPlane;

    for (int i = tid; i < TOH * TOW; i += NTHREADS) {
        const int oy = i >> 6, ox = i & 63;
        const int oh = oh0 + oy, ow = ow0 + ox;
        if (oh >= HO || ow >= WO) continue;
        const bool ro = (oh & 1) != 0;            // row parity -> kh taps
        const bool co = (ow & 1) != 0;            // col parity -> kw taps
        const int lyA = (ro ? ((oh + 1) >> 1) : (oh >> 1)) - r0;   // kh = ro?0:1
        const int lxA = (co ? ((ow + 1) >> 1) : (ow >> 1)) - c0;   // kw = co?0:1
        // weights by (kh,kw): static indices only -> stays in VGPRs
        const float wAA = ro ? (co ? swv[0] : swv[1]) : (co ? swv[3] : swv[4]);
        const float wAB = ro ? swv[2] : swv[5];   // (khA, 2), only if co
        const float wBA = co ? swv[6] : swv[7];   // (2, kwA), only if ro
        const float wBB = swv[8];                 // (2, 2),   only if ro&&co
        const int liA = lyA * TIW + lxA;
        float nom  = wAA * sNom[liA];
        float den  = wAA * sCf[liA];
        float cden = wAA;
        if (co) { nom += wAB * sNom[liA - 1];   den += wAB * sCf[liA - 1];   cden += wAB; }
        if (ro) {
            const int liB = liA - TIW;
            nom += wBA * sNom[liB];             den += wBA * sCf[liB];       cden += wBA;
            if (co) { nom += wBB * sNom[liB - 1]; den += wBB * sCf[liB - 1]; cden += wBB; }
        }
        const size_t o = (size_t)oh * WO + ow;
        // Outputs are write-once (218 MB total): non-temporal stores keep the
        // 110 MB input working set resident in the 192 MB L2 for halo reuse.
        __builtin_nontemporal_store((nom / (den + EPSC) + 2.0f * bs) * 0.5f, &gyOut[o]);
        __builtin_nontemporal_store(den / (cden + EPSC),                     &cgOut[o]);
    }
}

extern "C" void kernel_launch(void* const* d_in, const int* in_sizes, int n_in,
                              void* d_out, int out_size, void* d_ws, size_t ws_size,
                              hipStream_t stream) {
    (void)in_sizes; (void)n_in; (void)out_size; (void)d_ws; (void)ws_size;
    fused_structn_deconv<<<dim3(Bb * Cc * TILES_Y * TILES_X), dim3(NTHREADS), 0, stream>>>(
        (const float*)d_in[0],   // d
        (const float*)d_in[1],   // cd
        (const float*)d_in[2],   // gy
        (const float*)d_in[3],   // cgy
        (const float*)d_in[4],   // w_prop_raw
        (const float*)d_in[5],   // spatial_weight_raw
        (const float*)d_in[6],   // bias
        (float*)d_out);
}